// AggrEGATConv_38998303047882
// MI455X (gfx1250) — compile-verified
//
#include <hip/hip_runtime.h>
#include <hip/hip_bf16.h>

typedef __attribute__((ext_vector_type(2))) float v2f;
typedef __attribute__((ext_vector_type(8))) float v8f;

#define IN_N 128
#define IN_E 64
#define HDE 64   // H * OUT_E
#define H   4
#define DE  16
#define DN  16
#define ETILES 8 // 16-edge tiles per wave in k_edge (weight-fragment reuse)

// ---------------------------------------------------------------- utilities
__device__ inline void atomicMaxFloat(unsigned* addr, float v) {
    // IEEE ordering trick: non-negative floats compare as signed ints,
    // negative floats compare inverted as unsigned ints.
    if (v >= 0.0f) atomicMax((int*)addr, __float_as_int(v));
    else           atomicMin(addr, __float_as_uint(v));
}

// ---------------------------------------------------------------- K0: init
__global__ void k_init(unsigned* __restrict__ node_max,
                       float* __restrict__ node_sum,
                       float* __restrict__ res_n, int N) {
    int i = blockIdx.x * blockDim.x + threadIdx.x;
    if (i < N * H) { node_max[i] = 0xFF800000u; node_sum[i] = 0.0f; }
    if (i < N * DN) res_n[i] = 0.0f;
}

// ---------------------------------------------------------------- K1: node GEMMs
// One wave computes a 16-row tile of f_ni / f_nj / h_src  ([16 x 64] each),
// K = 128 in steps of 4 via V_WMMA_F32_16X16X4_F32. 12 v8f accumulators.
__global__ void k_node_gemm(const float* __restrict__ nf,
                            const float* __restrict__ Wni,
                            const float* __restrict__ Wnj,
                            const float* __restrict__ Wsrc,
                            const float* __restrict__ bsrc,
                            float* __restrict__ f_ni,
                            float* __restrict__ f_nj,
                            float* __restrict__ h_src, int N) {
    const int wave = (blockIdx.x * blockDim.x + threadIdx.x) >> 5;
    const int lane = threadIdx.x & 31;
    const int row0 = wave * 16;
    if (row0 >= N) return;
    const int half = lane >> 4;     // lanes 16..31 hold K+2/K+3 (A,B), M+8 (C)
    const int m    = lane & 15;

    v8f acc[3][4];
#pragma unroll
    for (int i = 0; i < 3; ++i)
#pragma unroll
        for (int t = 0; t < 4; ++t) acc[i][t] = v8f{0.f,0.f,0.f,0.f,0.f,0.f,0.f,0.f};

    int ra = row0 + m; if (ra >= N) ra = N - 1;
    const float* __restrict__ arow = nf + (size_t)ra * IN_N;

    for (int k0 = 0; k0 < IN_N; k0 += 4) {
        v2f a;
        a.x = arow[k0 + 2 * half];
        a.y = arow[k0 + 2 * half + 1];
#pragma unroll
        for (int mat = 0; mat < 3; ++mat) {
            const float* __restrict__ w = (mat == 0) ? Wni : (mat == 1) ? Wnj : Wsrc;
#pragma unroll
            for (int t = 0; t < 4; ++t) {
                const int col = t * 16 + m;
                v2f b;
                b.x = w[(k0 + 2 * half)     * HDE + col];
                b.y = w[(k0 + 2 * half + 1) * HDE + col];
                acc[mat][t] = __builtin_amdgcn_wmma_f32_16x16x4_f32(
                    false, a, false, b, (short)0, acc[mat][t], false, false);
            }
        }
    }

    const float bs0 = bsrc[0 * 16 + m], bs1 = bsrc[1 * 16 + m],
                bs2 = bsrc[2 * 16 + m], bs3 = bsrc[3 * 16 + m];
#pragma unroll
    for (int r = 0; r < 8; ++r) {
        const int row = row0 + r + 8 * half;
        if (row >= N) continue;
        const size_t o = (size_t)row * HDE;
#pragma unroll
        for (int t = 0; t < 4; ++t) {
            const int col = t * 16 + m;
            f_ni[o + col]  = acc[0][t][r];
            f_nj[o + col]  = acc[1][t][r];
        }
        h_src[o +  0 + m] = acc[2][0][r] + bs0;
        h_src[o + 16 + m] = acc[2][1][r] + bs1;
        h_src[o + 32 + m] = acc[2][2][r] + bs2;
        h_src[o + 48 + m] = acc[2][3][r] + bs3;
    }
}

// ---------------------------------------------------------------- K2: edge kernel
// Each wave preloads the complete W_fij WMMA fragment set into registers
// (16 k-steps x 4 N-tiles x v2f = 128 VGPRs) and then streams ETILES
// consecutive 16-edge tiles through it: steady state is one global_load_b64
// (efeats) + 4 v_wmma_f32_16x16x4_f32 per k-step — HBM-rate streaming.
// Epilogue fuses f_ni[src] + f_nj[dst] + bias_e, leaky_relu, res_e (head
// mean == mean of the 4 column tiles), attention logits (shfl_xor reduction
// within each 16-lane half) and the float atomic segment-max.
__global__ void k_edge(const float* __restrict__ ef,
                       const int*   __restrict__ src,
                       const int*   __restrict__ dst,
                       const float* __restrict__ Wfij,
                       const float* __restrict__ bias_e,
                       const float* __restrict__ attn,
                       const float* __restrict__ f_ni,
                       const float* __restrict__ f_nj,
                       float*    __restrict__ e_logits,
                       unsigned* __restrict__ node_max,
                       float*    __restrict__ res_e, int E, int N) {
    const int wave = (blockIdx.x * blockDim.x + threadIdx.x) >> 5;
    const int lane = threadIdx.x & 31;
    const int e_base = wave * 16 * ETILES;
    if (e_base >= E) return;
    const int half = lane >> 4;
    const int m    = lane & 15;

    // ---- preload all B (W_fij) fragments into registers
    v2f bw[16][4];
#pragma unroll
    for (int ks = 0; ks < 16; ++ks)
#pragma unroll
        for (int t = 0; t < 4; ++t) {
            const int col = t * 16 + m;
            bw[ks][t].x = Wfij[(4 * ks + 2 * half)     * HDE + col];
            bw[ks][t].y = Wfij[(4 * ks + 2 * half + 1) * HDE + col];
        }

    float bcol[4], acol[4];
#pragma unroll
    for (int t = 0; t < 4; ++t) {
        bcol[t] = bias_e[t * 16 + m];
        acol[t] = attn[t * 16 + m];
    }

    for (int tile = 0; tile < ETILES; ++tile) {
        const int e0 = e_base + tile * 16;
        if (e0 >= E) break;

        v8f acc[4];
#pragma unroll
        for (int t = 0; t < 4; ++t) acc[t] = v8f{0.f,0.f,0.f,0.f,0.f,0.f,0.f,0.f};

        int ra = e0 + m; if (ra >= E) ra = E - 1;
        const float* __restrict__ arow = ef + (size_t)ra * IN_E;
        if (e0 + 16 < E) __builtin_prefetch(ef + (size_t)(ra + 16) * IN_E, 0, 1);

#pragma unroll
        for (int ks = 0; ks < 16; ++ks) {
            v2f a;
            a.x = arow[4 * ks + 2 * half];
            a.y = arow[4 * ks + 2 * half + 1];
#pragma unroll
            for (int t = 0; t < 4; ++t)
                acc[t] = __builtin_amdgcn_wmma_f32_16x16x4_f32(
                    false, a, false, bw[ks][t], (short)0, acc[t], false, false);
        }

        int srci[8], dsti[8];
#pragma unroll
        for (int r = 0; r < 8; ++r) {
            int row = e0 + r + 8 * half; if (row >= E) row = E - 1;
            srci[r] = src[row];
            dsti[r] = dst[row];
        }

        float resv[8];
#pragma unroll
        for (int r = 0; r < 8; ++r) resv[r] = 0.0f;

#pragma unroll
        for (int t = 0; t < 4; ++t) {             // tile t == head h
            const int col = t * 16 + m;
            float lg[8];
#pragma unroll
            for (int r = 0; r < 8; ++r) {
                float v = acc[t][r]
                        + f_ni[(size_t)srci[r] * HDE + col]
                        + f_nj[(size_t)dsti[r] * HDE + col]
                        + bcol[t];
                v = v > 0.0f ? v : 0.01f * v;     // leaky_relu(0.01)
                resv[r] += v;
                lg[r] = v * acol[t];
            }
            // reduce over the 16 columns of this head (within each half-wave)
#pragma unroll
            for (int s = 1; s < 16; s <<= 1)
#pragma unroll
                for (int r = 0; r < 8; ++r) lg[r] += __shfl_xor(lg[r], s, 32);
            if (m == 0) {
#pragma unroll
                for (int r = 0; r < 8; ++r) {
                    const int row = e0 + r + 8 * half;
                    if (row < E) {
                        e_logits[(size_t)row * H + t] = lg[r];
                        atomicMaxFloat(node_max + (size_t)dsti[r] * H + t, lg[r]);
                    }
                }
            }
        }
#pragma unroll
        for (int r = 0; r < 8; ++r) {
            const int row = e0 + r + 8 * half;
            if (row < E) res_e[(size_t)row * DE + m] = 0.25f * resv[r];
        }
    }
}

// ---------------------------------------------------------------- K3: exp + segment sum
__global__ void k_softmax_sum(const int* __restrict__ dst,
                              float* __restrict__ e_logits,
                              const float* __restrict__ node_max,
                              float* __restrict__ node_sum, int E) {
    const int e = blockIdx.x * blockDim.x + threadIdx.x;
    if (e >= E) return;
    const int d = dst[e];
    float4 lv = ((const float4*)e_logits)[e];
    const float4 mv = ((const float4*)node_max)[d];
    float4 ex;
    ex.x = __expf(lv.x - mv.x);
    ex.y = __expf(lv.y - mv.y);
    ex.z = __expf(lv.z - mv.z);
    ex.w = __expf(lv.w - mv.w);
    ((float4*)e_logits)[e] = ex;
    atomicAdd(&node_sum[(size_t)d * H + 0], ex.x);
    atomicAdd(&node_sum[(size_t)d * H + 1], ex.y);
    atomicAdd(&node_sum[(size_t)d * H + 2], ex.z);
    atomicAdd(&node_sum[(size_t)d * H + 3], ex.w);
}

// ---------------------------------------------------------------- K4: aggregation
// One thread per (edge, d): res_n[dst][d] += (1/H) * sum_h a_h * h_src[src][h][d]
__global__ void k_aggregate(const int* __restrict__ src,
                            const int* __restrict__ dst,
                            const float* __restrict__ e_logits,
                            const float* __restrict__ node_sum,
                            const float* __restrict__ h_src,
                            float* __restrict__ res_n, int E) {
    const long long gid = (long long)blockIdx.x * blockDim.x + threadIdx.x;
    const int e = (int)(gid >> 4);
    const int d = (int)(gid & 15);
    if (e >= E) return;
    const int s = src[e], t = dst[e];
    const float4 ex = ((const float4*)e_logits)[e];
    const float4 z  = ((const float4*)node_sum)[t];
    const float* hs = h_src + (size_t)s * HDE + d;
    const float val = 0.25f * ((ex.x / z.x) * hs[0]
                             + (ex.y / z.y) * hs[16]
                             + (ex.z / z.z) * hs[32]
                             + (ex.w / z.w) * hs[48]);
    atomicAdd(&res_n[(size_t)t * DN + d], val);
}

// ---------------------------------------------------------------- launcher
extern "C" void kernel_launch(void* const* d_in, const int* in_sizes, int n_in,
                              void* d_out, int out_size, void* d_ws, size_t ws_size,
                              hipStream_t stream) {
    const float* nf     = (const float*)d_in[0];
    const float* ef     = (const float*)d_in[1];
    const int*   src    = (const int*)d_in[2];
    const int*   dst    = (const int*)d_in[3];
    const float* Wni    = (const float*)d_in[4];
    const float* Wnj    = (const float*)d_in[5];
    const float* Wfij   = (const float*)d_in[6];
    const float* Wsrc   = (const float*)d_in[7];
    const float* bsrc   = (const float*)d_in[8];
    const float* attn   = (const float*)d_in[9];
    const float* bias_e = (const float*)d_in[10];

    const int N = in_sizes[0] / IN_N;
    const int E = in_sizes[2];

    float* res_n = (float*)d_out;                     // [N, 16]
    float* res_e = (float*)d_out + (size_t)N * DN;    // [E, 16]

    // workspace carve-out (floats)
    float* f_ni     = (float*)d_ws;
    float* f_nj     = f_ni  + (size_t)N * HDE;
    float* h_src    = f_nj  + (size_t)N * HDE;
    float* e_logits = h_src + (size_t)N * HDE;
    unsigned* node_max = (unsigned*)(e_logits + (size_t)E * H);
    float*    node_sum = (float*)(node_max + (size_t)N * H);

    // K0: init
    {
        int n = N * DN;
        k_init<<<(n + 255) / 256, 256, 0, stream>>>(node_max, node_sum, res_n, N);
    }
    // K1: node GEMMs (1 wave per 16-row tile)
    {
        int waves = (N + 15) / 16;
        int blocks = (waves * 32 + 255) / 256;
        k_node_gemm<<<blocks, 256, 0, stream>>>(nf, Wni, Wnj, Wsrc, bsrc,
                                                f_ni, f_nj, h_src, N);
    }
    // K2: edge GEMM + softmax-max (1 wave per ETILES*16 edges, weights in regs)
    {
        int waves = (E + 16 * ETILES - 1) / (16 * ETILES);
        int blocks = (waves * 32 + 255) / 256;
        k_edge<<<blocks, 256, 0, stream>>>(ef, src, dst, Wfij, bias_e, attn,
                                           f_ni, f_nj, e_logits, node_max, res_e,
                                           E, N);
    }
    // K3: exp + segment sum
    k_softmax_sum<<<(E + 255) / 256, 256, 0, stream>>>(dst, e_logits, (const float*)node_max,
                                                       node_sum, E);
    // K4: aggregate into res_n
    {
        long long n = (long long)E * DN;
        int blocks = (int)((n + 255) / 256);
        k_aggregate<<<blocks, 256, 0, stream>>>(src, dst, e_logits, node_sum,
                                                h_src, res_n, E);
    }
}